// IGMC_15590731284646
// MI455X (gfx1250) — compile-verified
//
#include <hip/hip_runtime.h>
#include <math.h>

typedef __attribute__((ext_vector_type(2))) float v2f;
typedef __attribute__((ext_vector_type(8))) float v8f;

#define RREL 5
#define NBAS 2
#define DOUT 32

// ---------------- count / inverse-mean norm (edge structure is layer-invariant) ----------------
__global__ void k_edge_count(const int* __restrict__ dst, const int* __restrict__ et,
                             float* __restrict__ cnt, int E) {
  int e = blockIdx.x * blockDim.x + threadIdx.x;
  if (e < E) atomicAdd(&cnt[(size_t)dst[e] * RREL + et[e]], 1.0f);
}

__global__ void k_inv_norm(float* __restrict__ cnt, int n) {
  int i = blockIdx.x * blockDim.x + threadIdx.x;
  if (i < n) cnt[i] = 1.0f / fmaxf(cnt[i], 1.0f);
}

// ------------- per-edge aggregate-first scatter: s[(dst*R+et)*IN + j] += h[src*IN + j] -------------
template <int LOGIN>
__global__ void k_edge_scatter(const int* __restrict__ src, const int* __restrict__ dst,
                               const int* __restrict__ et, const float* __restrict__ hin,
                               float* __restrict__ s, long long total) {
  long long t = (long long)blockIdx.x * blockDim.x + threadIdx.x;
  if (t >= total) return;
  int e = (int)(t >> LOGIN);
  int j = (int)(t & ((1 << LOGIN) - 1));
  int sn = src[e];
  int dn = dst[e];
  int r  = et[e];
  float v = hin[((size_t)sn << LOGIN) + j];
  atomicAdd(&s[(((size_t)dn * RREL + r) << LOGIN) + j], v);
}

// ------------- dense transform: h = tanh( [s_norm | x] @ [Wcat | root] + bias ), WMMA f32 -------------
// One wave32 per 16-node tile. Wcat = { W[r] = comp[r]@basis } for r<5, then root. K_total = 6*IN.
// A 16x4 f32 layout: lanes 0-15 -> M=row, V0=K0,V1=K1; lanes 16-31 -> V0=K2,V1=K3.
// B 4x16  f32 layout (symmetric over N); C/D: VGPR v -> row v (lanes 0-15) / row v+8 (lanes 16-31).
template <int IN>
__global__ void k_rgcn_transform(const float* __restrict__ s, const float* __restrict__ norm,
                                 const float* __restrict__ xin, const float* __restrict__ comp,
                                 const float* __restrict__ basis, const float* __restrict__ root,
                                 const float* __restrict__ bias, float* __restrict__ hout, int N) {
  __shared__ float Wlds[(RREL + 1) * IN * DOUT];
  __shared__ float blds[DOUT];
  int tid = threadIdx.x;
  for (int idx = tid; idx < RREL * IN * DOUT; idx += blockDim.x) {
    int r = idx / (IN * DOUT);
    int rem = idx - r * (IN * DOUT);
    float w = 0.f;
#pragma unroll
    for (int nb = 0; nb < NBAS; ++nb)
      w += comp[r * NBAS + nb] * basis[nb * IN * DOUT + rem];
    Wlds[idx] = w;
  }
  for (int idx = tid; idx < IN * DOUT; idx += blockDim.x)
    Wlds[RREL * IN * DOUT + idx] = root[idx];
  if (tid < DOUT) blds[tid] = bias[tid];
  __syncthreads();

  int lane = tid & 31;
  int wave = tid >> 5;
  int tile = blockIdx.x * (blockDim.x >> 5) + wave;
  int ntiles = N >> 4;
  if (tile >= ntiles) return;          // wave-uniform: EXEC stays all-ones for WMMA

  int row = (tile << 4) + (lane & 15);
  int kp  = (lane >> 4) << 1;          // 0 (lanes 0-15) or 2 (lanes 16-31)
  int col = lane & 15;

  v8f acc0 = {};
  v8f acc1 = {};

  for (int r = 0; r < RREL; ++r) {
    float nm = norm[(size_t)row * RREL + r];
    const float* srow = s + ((size_t)row * RREL + r) * IN;
    const float* wrow = Wlds + r * IN * DOUT;
#pragma unroll
    for (int kc = 0; kc < IN; kc += 4) {
      v2f a  = { srow[kc + kp] * nm, srow[kc + kp + 1] * nm };
      v2f b0 = { wrow[(kc + kp) * DOUT + col],      wrow[(kc + kp + 1) * DOUT + col] };
      v2f b1 = { wrow[(kc + kp) * DOUT + 16 + col], wrow[(kc + kp + 1) * DOUT + 16 + col] };
      acc0 = __builtin_amdgcn_wmma_f32_16x16x4_f32(false, a, false, b0, (short)0, acc0, false, false);
      acc1 = __builtin_amdgcn_wmma_f32_16x16x4_f32(false, a, false, b1, (short)0, acc1, false, false);
    }
  }
  {
    const float* xrow = xin + (size_t)row * IN;
    const float* wrow = Wlds + RREL * IN * DOUT;
#pragma unroll
    for (int kc = 0; kc < IN; kc += 4) {
      v2f a  = { xrow[kc + kp], xrow[kc + kp + 1] };
      v2f b0 = { wrow[(kc + kp) * DOUT + col],      wrow[(kc + kp + 1) * DOUT + col] };
      v2f b1 = { wrow[(kc + kp) * DOUT + 16 + col], wrow[(kc + kp + 1) * DOUT + 16 + col] };
      acc0 = __builtin_amdgcn_wmma_f32_16x16x4_f32(false, a, false, b0, (short)0, acc0, false, false);
      acc1 = __builtin_amdgcn_wmma_f32_16x16x4_f32(false, a, false, b1, (short)0, acc1, false, false);
    }
  }

  int mofs = (lane >> 4) << 3;         // 0 or 8
#pragma unroll
  for (int v = 0; v < 8; ++v) {
    int node = (tile << 4) + mofs + v;
    hout[(size_t)node * DOUT + col]      = tanhf(acc0[v] + blds[col]);
    hout[(size_t)node * DOUT + 16 + col] = tanhf(acc1[v] + blds[16 + col]);
  }
}

// scalar fallback for nodes past the last full 16-tile (unused when 16 | N)
template <int IN>
__global__ void k_rgcn_tail(const float* __restrict__ s, const float* __restrict__ norm,
                            const float* __restrict__ xin, const float* __restrict__ comp,
                            const float* __restrict__ basis, const float* __restrict__ root,
                            const float* __restrict__ bias, float* __restrict__ hout,
                            int n0, int N) {
  int n = n0 + blockIdx.x;
  if (n >= N) return;
  int o = threadIdx.x;                 // 32 threads, one output each
  float acc = bias[o];
  for (int r = 0; r < RREL; ++r) {
    float nm = norm[(size_t)n * RREL + r];
    for (int i = 0; i < IN; ++i) {
      float w = 0.f;
      for (int nb = 0; nb < NBAS; ++nb) w += comp[r * NBAS + nb] * basis[(nb * IN + i) * DOUT + o];
      acc += s[((size_t)n * RREL + r) * IN + i] * nm * w;
    }
  }
  for (int i = 0; i < IN; ++i) acc += xin[(size_t)n * IN + i] * root[i * DOUT + o];
  hout[(size_t)n * DOUT + o] = tanhf(acc);
}

// ---------------- readout: sel[b, lofs:lofs+32] = h[start[b]] ----------------
__global__ void k_select(const float* __restrict__ h, const int* __restrict__ startn,
                         float* __restrict__ sel, int lofs) {
  int b = blockIdx.x;
  int o = threadIdx.x;                 // 32
  sel[(size_t)b * 128 + lofs + o] = h[(size_t)startn[b] * DOUT + o];
}

// ---------------- MLP head + log_softmax (one block of 128 per graph row) ----------------
__global__ void k_mlp_head(const float* __restrict__ sel, const float* __restrict__ w1,
                           const float* __restrict__ b1, const float* __restrict__ w2,
                           const float* __restrict__ b2, float* __restrict__ out, int C) {
  __shared__ float row[128];
  __shared__ float h1[128];
  __shared__ float lg[16];
  int b = blockIdx.x;
  int t = threadIdx.x;
  row[t] = sel[(size_t)b * 128 + t];
  __syncthreads();
  float acc = b1[t];
  for (int i = 0; i < 128; ++i) acc += row[i] * w1[i * 128 + t];
  h1[t] = fmaxf(acc, 0.0f);
  __syncthreads();
  if (t < C) {
    float a = b2[t];
    for (int i = 0; i < 128; ++i) a += h1[i] * w2[i * C + t];
    lg[t] = a;
  }
  __syncthreads();
  if (t == 0) {
    float m = lg[0];
    for (int c = 1; c < C; ++c) m = fmaxf(m, lg[c]);
    float ssum = 0.f;
    for (int c = 0; c < C; ++c) ssum += expf(lg[c] - m);
    float lse = m + logf(ssum);
    for (int c = 0; c < C; ++c) out[(size_t)b * C + c] = lg[c] - lse;
  }
}

extern "C" void kernel_launch(void* const* d_in, const int* in_sizes, int n_in,
                              void* d_out, int out_size, void* d_ws, size_t ws_size,
                              hipStream_t stream) {
  const float* x      = (const float*)d_in[0];
  const int*   ei     = (const int*)d_in[1];
  const int*   et     = (const int*)d_in[2];
  const int*   startn = (const int*)d_in[3];
  const float* basis0 = (const float*)d_in[4];
  const float* comp0  = (const float*)d_in[5];
  const float* root0  = (const float*)d_in[6];
  const float* bias0  = (const float*)d_in[7];
  const float* basis  = (const float*)d_in[8];
  const float* comp   = (const float*)d_in[9];
  const float* root   = (const float*)d_in[10];
  const float* bias   = (const float*)d_in[11];
  const float* w1     = (const float*)d_in[12];
  const float* b1     = (const float*)d_in[13];
  const float* w2     = (const float*)d_in[14];
  const float* b2     = (const float*)d_in[15];
  float* out = (float*)d_out;
  (void)n_in; (void)ws_size;

  const int N = in_sizes[0] / 4;       // F = 4
  const int E = in_sizes[1] / 2;
  const int B = in_sizes[3];
  const int C = in_sizes[15];

  const int* src = ei;
  const int* dst = ei + E;

  // workspace carve-out (256B aligned): norm | s | hA | hB | sel  (~92 MB total)
  char* ws = (char*)d_ws;
  size_t off = 0;
  auto carve = [&](size_t bytes) -> char* {
    char* p = ws + off;
    off = (off + bytes + 255) & ~(size_t)255;
    return p;
  };
  float* norm = (float*)carve((size_t)N * RREL * sizeof(float));
  float* sbuf = (float*)carve((size_t)N * RREL * DOUT * sizeof(float));
  float* hA   = (float*)carve((size_t)N * DOUT * sizeof(float));
  float* hB   = (float*)carve((size_t)N * DOUT * sizeof(float));
  float* sel  = (float*)carve((size_t)B * 128 * sizeof(float));

  // per-(dst, relation) inverse mean counts (edge-structure invariant across layers)
  hipMemsetAsync(norm, 0, (size_t)N * RREL * sizeof(float), stream);
  k_edge_count<<<(E + 255) / 256, 256, 0, stream>>>(dst, et, norm, E);
  k_inv_norm<<<((N * RREL) + 255) / 256, 256, 0, stream>>>(norm, N * RREL);

  const int ntiles  = N >> 4;
  const int tail    = N - (ntiles << 4);
  const int tblocks = (ntiles + 7) / 8;    // 8 waves (16-node tiles) per 256-thread block

  // ---- layer 0: IN = 4 ----
  hipMemsetAsync(sbuf, 0, (size_t)N * RREL * 4 * sizeof(float), stream);
  {
    long long tot = (long long)E << 2;
    k_edge_scatter<2><<<(unsigned)((tot + 255) / 256), 256, 0, stream>>>(src, dst, et, x, sbuf, tot);
  }
  k_rgcn_transform<4><<<tblocks, 256, 0, stream>>>(sbuf, norm, x, comp0, basis0, root0, bias0, hA, N);
  if (tail) k_rgcn_tail<4><<<tail, 32, 0, stream>>>(sbuf, norm, x, comp0, basis0, root0, bias0, hA,
                                                    ntiles << 4, N);
  k_select<<<B, 32, 0, stream>>>(hA, startn, sel, 0);

  // ---- layers 1..3: IN = 32 ----
  float* hp = hA;
  float* hc = hB;
  for (int l = 0; l < 3; ++l) {
    hipMemsetAsync(sbuf, 0, (size_t)N * RREL * DOUT * sizeof(float), stream);
    long long tot = (long long)E << 5;
    k_edge_scatter<5><<<(unsigned)((tot + 255) / 256), 256, 0, stream>>>(src, dst, et, hp, sbuf, tot);
    const float* cl = comp + l * RREL * NBAS;
    const float* bl = basis + (size_t)l * NBAS * DOUT * DOUT;
    const float* rl = root + (size_t)l * DOUT * DOUT;
    const float* il = bias + l * DOUT;
    k_rgcn_transform<32><<<tblocks, 256, 0, stream>>>(sbuf, norm, hp, cl, bl, rl, il, hc, N);
    if (tail) k_rgcn_tail<32><<<tail, 32, 0, stream>>>(sbuf, norm, hp, cl, bl, rl, il, hc,
                                                       ntiles << 4, N);
    k_select<<<B, 32, 0, stream>>>(hc, startn, sel, (l + 1) * DOUT);
    float* tmp = hp; hp = hc; hc = tmp;
  }

  k_mlp_head<<<B, 128, 0, stream>>>(sel, w1, b1, w2, b2, out, C);
}